// Net_54803782697308
// MI455X (gfx1250) — compile-verified
//
#include <hip/hip_runtime.h>

typedef float v2f __attribute__((ext_vector_type(2)));
typedef float v8f __attribute__((ext_vector_type(8)));
typedef int   v4i __attribute__((ext_vector_type(4)));

#define IN_F  256
#define MID_F 128
#define OUT_F 64

#define AS1 __attribute__((address_space(1)))
#define AS3 __attribute__((address_space(3)))

#if defined(__AMDGCN__) && __has_builtin(__builtin_amdgcn_global_load_async_to_lds_b128)
#define HAVE_ASYNC_LDS 1
#else
#define HAVE_ASYNC_LDS 0
#endif

__device__ __forceinline__ void wait_async_then_barrier() {
#if HAVE_ASYNC_LDS
#if __has_builtin(__builtin_amdgcn_s_wait_asynccnt)
    __builtin_amdgcn_s_wait_asynccnt(0);
#else
    asm volatile("s_wait_asynccnt 0" ::: "memory");
#endif
#endif
    __syncthreads();
}

// ---------------------------------------------------------------------------
// Degree / normalization kernels
// ---------------------------------------------------------------------------
__global__ void k_init_deg(float* __restrict__ deg, int n) {
    int i = blockIdx.x * blockDim.x + threadIdx.x;
    if (i < n) deg[i] = 1.0f;  // self-loop contributes 1 to every node's degree
}

__global__ void k_count_deg(const long long* __restrict__ dst, float* __restrict__ deg, int e) {
    int i = blockIdx.x * blockDim.x + threadIdx.x;
    if (i < e) atomicAdd(&deg[(int)dst[i]], 1.0f);
}

__global__ void k_make_dinv(float* __restrict__ deg, int n) {
    int i = blockIdx.x * blockDim.x + threadIdx.x;
    if (i < n) {
        float d = deg[i];
        deg[i] = (d > 0.0f) ? __frsqrt_rn(d) : 0.0f;  // deg >= 1 always here
    }
}

// ---------------------------------------------------------------------------
// fp32 WMMA GEMM:  D[M,N] = A[M,K] @ W[K,N]      (M % 32 == 0)
// Block = N/16 waves; each wave owns one 16-col tile and TWO 16-row tiles
// (rows row0..+15 and row0+16..+31) so a single B fragment feeds two
// V_WMMA_F32_16X16X4_F32 issues. 32-row A strip staged in LDS (padded
// stride), via GLOBAL_LOAD_ASYNC_TO_LDS_B128 when the toolchain exposes it.
//
// A-fragment (16x4 f32, ISA 7.12.2): lanes 0-15 hold M=lane, VGPR0=K0,
// VGPR1=K0+1; lanes 16-31 hold M=lane-16, VGPR0=K0+2, VGPR1=K0+3.
// B-fragment (4x16): lane%16 = N; VGPR0 = row K0 (lanes 0-15) / K0+2 (16-31).
// C/D (16x16 f32): VGPR v = row v + 8*(lane>=16), col = lane%16.
// ---------------------------------------------------------------------------
template <int K, int N>
__global__ __launch_bounds__(32 * (N / 16))
void k_gemm_wmma_f32(const float* __restrict__ A, const float* __restrict__ W,
                     float* __restrict__ D) {
    constexpr int WAVES = N / 16;
    constexpr int LDA   = K + 4;                 // pad: kills LDS bank conflicts
    __shared__ float As[32 * LDA];

    const int row0 = blockIdx.x * 32;
    const int tid  = threadIdx.x;

    // Stage the 32 x K A-strip into LDS (16B chunks; pad keeps them aligned).
    constexpr int CPR = K / 4;                   // float4 chunks per row
    for (int i = tid; i < 32 * CPR; i += WAVES * 32) {
        int r = i / CPR, c4 = (i - r * CPR) * 4;
        const float* g = A + (size_t)(row0 + r) * K + c4;
        float*       l = As + r * LDA + c4;
#if HAVE_ASYNC_LDS
        __builtin_amdgcn_global_load_async_to_lds_b128(
            (AS1 v4i*)const_cast<float*>(g), (AS3 v4i*)l, 0, 0);
#else
        *reinterpret_cast<float4*>(l) = *reinterpret_cast<const float4*>(g);
#endif
    }
    wait_async_then_barrier();

    const int wave = tid >> 5;
    const int lane = tid & 31;
    const int hw   = lane >> 4;                  // half-wave select
    const int l16  = lane & 15;
    const int col0 = wave * 16;

    const float* as0 = As + l16 * LDA;           // rows  0..15 of strip
    const float* as1 = As + (16 + l16) * LDA;    // rows 16..31 of strip
    const float* wp  = W + col0 + l16;

    v8f acc0 = {}, acc1 = {};
#pragma unroll 8
    for (int k0 = 0; k0 < K; k0 += 4) {
        const int kk = k0 + 2 * hw;
        v2f a0, a1, b;
        a0.x = as0[kk + 0];  a0.y = as0[kk + 1];
        a1.x = as1[kk + 0];  a1.y = as1[kk + 1];
        b.x  = wp[(size_t)(kk + 0) * N];
        b.y  = wp[(size_t)(kk + 1) * N];
        acc0 = __builtin_amdgcn_wmma_f32_16x16x4_f32(false, a0, false, b,
                                                     (short)0, acc0, false, false);
        acc1 = __builtin_amdgcn_wmma_f32_16x16x4_f32(false, a1, false, b,
                                                     (short)0, acc1, false, false);
    }

    // Branch-free stores: M is a multiple of 32, no tail.
    float* d0 = D + (size_t)(row0 + 8 * hw) * N + col0 + l16;
    float* d1 = d0 + (size_t)16 * N;
#pragma unroll
    for (int v = 0; v < 8; ++v) d0[(size_t)v * N] = acc0[v];
#pragma unroll
    for (int v = 0; v < 8; ++v) d1[(size_t)v * N] = acc1[v];
}

// ---------------------------------------------------------------------------
// Aggregation: init with bias + self-loop term, then per-edge scatter-add.
// ---------------------------------------------------------------------------
template <int F>
__global__ void k_init_self(const float* __restrict__ h, const float* __restrict__ bias,
                            const float* __restrict__ dinv, float* __restrict__ agg, int n) {
    int i = blockIdx.x * blockDim.x + threadIdx.x;     // over n * F
    if (i >= n * F) return;
    int node = i / F, f = i - node * F;
    float w = dinv[node];
    agg[i] = bias[f] + h[i] * (w * w);                  // self-loop norm = dinv^2
}

// F/4 lanes cooperate per edge (F=128 -> one full wave per edge).
template <int F>
__global__ void k_edge_agg(const float* __restrict__ h,
                           const long long* __restrict__ src,
                           const long long* __restrict__ dst,
                           const float* __restrict__ dinv,
                           float* __restrict__ agg, int e) {
    constexpr int CH = F / 4;                           // float4 chunks per edge
    long long t = (long long)blockIdx.x * blockDim.x + threadIdx.x;
    long long edge = t / CH;
    int ch = (int)(t - edge * CH);
    if (edge >= e) return;

    size_t s = (size_t)src[edge];
    size_t d = (size_t)dst[edge];
    float  w = dinv[s] * dinv[d];

    const float4 v = *reinterpret_cast<const float4*>(h + s * F + 4 * ch);
    float* o = agg + d * F + 4 * ch;
    atomicAdd(o + 0, v.x * w);
    atomicAdd(o + 1, v.y * w);
    atomicAdd(o + 2, v.z * w);
    atomicAdd(o + 3, v.w * w);
}

__global__ void k_relu(float* __restrict__ a, long long n) {
    long long i = (long long)blockIdx.x * blockDim.x + threadIdx.x;
    if (i < n) a[i] = fmaxf(a[i], 0.0f);
}

// ---------------------------------------------------------------------------
// Launch
// ---------------------------------------------------------------------------
extern "C" void kernel_launch(void* const* d_in, const int* in_sizes, int n_in,
                              void* d_out, int out_size, void* d_ws, size_t ws_size,
                              hipStream_t stream) {
    const float*     x  = (const float*)d_in[0];
    const long long* ei = (const long long*)d_in[1];
    const float*     W1 = (const float*)d_in[2];
    const float*     b1 = (const float*)d_in[3];
    const float*     W2 = (const float*)d_in[4];
    const float*     b2 = (const float*)d_in[5];
    float*           out = (float*)d_out;

    const int N = in_sizes[0] / IN_F;      // 100000 nodes (multiple of 32)
    const int E = in_sizes[1] / 2;         // 1600000 edges
    const long long* src = ei;
    const long long* dst = ei + E;

    // Workspace carve-up (256B aligned).
    char* p = (char*)d_ws;
    auto carve = [&](size_t bytes) {
        char* q = p;
        p += (bytes + 255) & ~(size_t)255;
        return q;
    };
    float* dinv = (float*)carve((size_t)N * sizeof(float));
    float* h1   = (float*)carve((size_t)N * MID_F * sizeof(float));
    float* agg1 = (float*)carve((size_t)N * MID_F * sizeof(float));
    float* h2   = (float*)carve((size_t)N * OUT_F * sizeof(float));

    const int T = 256;

    // Degrees -> dinv
    k_init_deg<<<(N + T - 1) / T, T, 0, stream>>>(dinv, N);
    k_count_deg<<<(E + T - 1) / T, T, 0, stream>>>(dst, dinv, E);
    k_make_dinv<<<(N + T - 1) / T, T, 0, stream>>>(dinv, N);

    // Layer 1: h1 = x @ W1
    k_gemm_wmma_f32<IN_F, MID_F><<<N / 32, 32 * (MID_F / 16), 0, stream>>>(x, W1, h1);
    k_init_self<MID_F><<<((long long)N * MID_F + T - 1) / T, T, 0, stream>>>(h1, b1, dinv, agg1, N);
    {
        long long total = (long long)E * (MID_F / 4);
        k_edge_agg<MID_F><<<(total + T - 1) / T, T, 0, stream>>>(h1, src, dst, dinv, agg1, E);
    }
    k_relu<<<((long long)N * MID_F + T - 1) / T, T, 0, stream>>>(agg1, (long long)N * MID_F);

    // Layer 2: h2 = relu(agg1) @ W2
    k_gemm_wmma_f32<MID_F, OUT_F><<<N / 32, 32 * (OUT_F / 16), 0, stream>>>(agg1, W2, h2);
    k_init_self<OUT_F><<<((long long)N * OUT_F + T - 1) / T, T, 0, stream>>>(h2, b2, dinv, out, N);
    {
        long long total = (long long)E * (OUT_F / 4);
        k_edge_agg<OUT_F><<<(total + T - 1) / T, T, 0, stream>>>(h2, src, dst, dinv, out, E);
    }
}